// EquivariantMPNNEmbedding_42374147342442
// MI455X (gfx1250) — compile-verified
//
#include <hip/hip_runtime.h>
#include <math.h>

#define B_   128
#define NEL  32
#define NION 8
#define HCH  128

typedef float v2f __attribute__((ext_vector_type(2)));
typedef float v8f __attribute__((ext_vector_type(8)));

// V_WMMA_F32_16X16X4_F32 : D(16x16 f32) = A(16x4 f32) * B(4x16 f32) + C
__device__ inline v8f wmma4(v2f a, v2f b, v8f c) {
  return __builtin_amdgcn_wmma_f32_16x16x4_f32(false, a, false, b, (short)0, c, false, false);
}

__device__ inline float siluf(float x) { return x / (1.0f + __expf(-x)); }
__device__ inline float geluf(float x) {        // JAX default: tanh-approx gelu
  const float k = 0.7978845608028654f;
  return 0.5f * x * (1.0f + tanhf(k * (x + 0.044715f * x * x * x)));
}

// ---------------------------------------------------------------------------
// Generic strided WMMA GEMM:  D[m, n*sD] = sum_k A[m*ldA + k*sA] * W[k*N + n]
// M = 4096 rows (B*NEL), grid = (M/16, N/64), block = 128 (4 waves, 16x16/wave)
// ---------------------------------------------------------------------------
__global__ __launch_bounds__(128)
void gemm_wmma_k(const float* __restrict__ A, int ldA, int sA,
                 const float* __restrict__ W, int N,
                 float* __restrict__ D, int ldD, int sD, int K)
{
  const int wave = threadIdx.x >> 5;
  const int lane = threadIdx.x & 31;
  const int m0 = blockIdx.x << 4;
  const int n0 = (blockIdx.y << 6) + (wave << 4);
  if (n0 >= N) return;                 // wave-uniform
  const int r    = lane & 15;          // A row in tile == B col in tile
  const int half = lane >> 4;
  const int h2   = half << 1;

  v8f acc = {0.f,0.f,0.f,0.f,0.f,0.f,0.f,0.f};
  const float* Arow = A + (size_t)(m0 + r) * ldA;
  const float* Wcol = W + n0 + r;
  for (int k = 0; k < K; k += 4) {
    v2f a, b;
    a.x = Arow[(k + h2) * sA];
    a.y = Arow[(k + h2 + 1) * sA];
    b.x = Wcol[(size_t)(k + h2) * N];
    b.y = Wcol[(size_t)(k + h2 + 1) * N];
    acc = wmma4(a, b, acc);
  }
  float* Drow = D + (size_t)(m0 + (half << 3)) * ldD + (size_t)(n0 + r) * sD;
  #pragma unroll
  for (int v = 0; v < 8; v++) Drow[(size_t)v * ldD] = acc[v];
}

// ---------------------------------------------------------------------------
// Fused radial-MLP [32 ->64 ->64 -> nw] + equivariant interact + scatter-add.
// One block = 32 edges of one batch; 8 waves; weights + activations in LDS.
// ---------------------------------------------------------------------------
__global__ __launch_bounds__(256)
void radint_k(const float* __restrict__ featE, int srcDim,
              const float* __restrict__ diffE, const float* __restrict__ distE,
              const int* __restrict__ srcIdx, const int* __restrict__ tgtIdx,
              const float* __restrict__ W1, const float* __restrict__ b1,
              const float* __restrict__ W2, const float* __restrict__ b2,
              const float* __restrict__ W3, const float* __restrict__ b3, int nw,
              const float* __restrict__ srcS, int srcBatched,
              const float* __restrict__ srcV,      // nullptr when no vector feats
              float* __restrict__ catS, int offS,
              float* __restrict__ catV, int offV)
{
  __shared__ float sW1[32 * 64];
  __shared__ float sW2[64 * 64];
  __shared__ float sW3[64 * 256];
  __shared__ float sB1[64];
  __shared__ float sB2[64];
  __shared__ float sB3[256];
  __shared__ float sX [32 * 33];   // padded pitch -> conflict-free A reads
  __shared__ float sH1[32 * 65];
  __shared__ float sH2[32 * 65];
  __shared__ float sWo[32 * 256];
  __shared__ float sY [32 * 4];    // Y0, Y1x, Y1y, Y1z
  __shared__ int   sS[32];
  __shared__ int   sT[32];

  const int t  = threadIdx.x;
  const int b  = blockIdx.y;
  const int e0 = blockIdx.x << 5;

  for (int i = t; i < 32 * 64; i += 256) sW1[i] = W1[i];
  for (int i = t; i < 64 * 64; i += 256) sW2[i] = W2[i];
  for (int i = t; i < 64 * nw; i += 256) sW3[i] = W3[i];
  if (t < 64) { sB1[t] = b1[t]; sB2[t] = b2[t]; }
  for (int i = t; i < nw; i += 256) sB3[i] = b3[i];

  if (t < 32) {                                   // edge meta + spherical terms
    int e = e0 + t;
    int s = srcIdx[e], tg = tgtIdx[e];
    sS[t] = s; sT[t] = tg;
    int po = (b * NEL + tg) * srcDim + s;
    float d   = distE[po];
    float inv = d > 0.f ? 1.f / fmaxf(d, 1e-12f) : 0.f;
    float fl  = __cosf(1.57079632679f * fminf(d * 0.2f, 1.f));   // R_LONG = 5
    float sc  = 1.7320508075688772f * inv * tanhf(d * 2.f) * fl; // R_SHORT=0.5
    sY[t * 4 + 0] = fl;
    sY[t * 4 + 1] = diffE[po * 3 + 0] * sc;
    sY[t * 4 + 2] = diffE[po * 3 + 1] * sc;
    sY[t * 4 + 3] = diffE[po * 3 + 2] * sc;
  }
  for (int i = t; i < 32 * 32; i += 256) {        // gather edge features
    int e = i >> 5, k = i & 31;
    int s = srcIdx[e0 + e], tg = tgtIdx[e0 + e];
    sX[e * 33 + k] = featE[((size_t)(b * NEL + tg) * srcDim + s) * 32 + k];
  }
  __syncthreads();

  const int wave = t >> 5, lane = t & 31;
  const int rt = wave >> 2, ct = wave & 3;
  const int r = lane & 15, half = lane >> 4, h2 = half << 1;
  const int row0 = rt << 4;

  { // layer 1: X(32x32) @ W1(32x64), silu
    v8f acc = {0.f,0.f,0.f,0.f,0.f,0.f,0.f,0.f};
    #pragma unroll
    for (int k = 0; k < 32; k += 4) {
      v2f a, bv;
      a.x  = sX[(row0 + r) * 33 + k + h2];
      a.y  = sX[(row0 + r) * 33 + k + h2 + 1];
      bv.x = sW1[(k + h2) * 64 + (ct << 4) + r];
      bv.y = sW1[(k + h2 + 1) * 64 + (ct << 4) + r];
      acc = wmma4(a, bv, acc);
    }
    float bb = sB1[(ct << 4) + r];
    #pragma unroll
    for (int v = 0; v < 8; v++)
      sH1[(row0 + v + (half << 3)) * 65 + (ct << 4) + r] = siluf(acc[v] + bb);
  }
  __syncthreads();

  { // layer 2: H1 @ W2(64x64), silu
    v8f acc = {0.f,0.f,0.f,0.f,0.f,0.f,0.f,0.f};
    #pragma unroll
    for (int k = 0; k < 64; k += 4) {
      v2f a, bv;
      a.x  = sH1[(row0 + r) * 65 + k + h2];
      a.y  = sH1[(row0 + r) * 65 + k + h2 + 1];
      bv.x = sW2[(k + h2) * 64 + (ct << 4) + r];
      bv.y = sW2[(k + h2 + 1) * 64 + (ct << 4) + r];
      acc = wmma4(a, bv, acc);
    }
    float bb = sB2[(ct << 4) + r];
    #pragma unroll
    for (int v = 0; v < 8; v++)
      sH2[(row0 + v + (half << 3)) * 65 + (ct << 4) + r] = siluf(acc[v] + bb);
  }
  __syncthreads();

  { // layer 3: H2 @ W3(64 x nw), bias only
    const int ntile = nw >> 4;
    for (int c3 = ct; c3 < ntile; c3 += 4) {
      v8f acc = {0.f,0.f,0.f,0.f,0.f,0.f,0.f,0.f};
      #pragma unroll
      for (int k = 0; k < 64; k += 4) {
        v2f a, bv;
        a.x  = sH2[(row0 + r) * 65 + k + h2];
        a.y  = sH2[(row0 + r) * 65 + k + h2 + 1];
        bv.x = sW3[(k + h2) * nw + (c3 << 4) + r];
        bv.y = sW3[(k + h2 + 1) * nw + (c3 << 4) + r];
        acc = wmma4(a, bv, acc);
      }
      float bb = sB3[(c3 << 4) + r];
      #pragma unroll
      for (int v = 0; v < 8; v++)
        sWo[(row0 + v + (half << 3)) * nw + (c3 << 4) + r] = acc[v] + bb;
    }
  }
  __syncthreads();

  // interact: thread -> (channel c, component comp); comp is wave-uniform.
  const int c = t & 63;
  const int comp = t >> 6;                 // 0 = scalar msg, 1..3 = vector msg
  float acc = 0.f;
  int curT = sT[0];
  for (int e = 0; e < 32; e++) {
    int tg = sT[e];
    if (tg != curT) {                      // edges are target-contiguous
      if (comp == 0) atomicAdd(&catS[(b * NEL + curT) * 320 + offS + c], acc);
      else atomicAdd(&catV[((b * NEL + curT) * 320 + offV + c) * 3 + (comp - 1)], acc);
      acc = 0.f; curT = tg;
    }
    int s = sS[e];
    float ss = srcBatched ? srcS[(b * NEL + s) * 64 + c] : srcS[s * 64 + c];
    const float* wr = &sWo[e * nw];
    float Y0 = sY[e * 4];
    if (comp == 0) {
      float m = wr[c] * ss * Y0;
      if (srcV) {
        const float* vp = &srcV[((b * NEL + s) * 64 + c) * 3];
        m += wr[128 + c] * (vp[0] * sY[e*4+1] + vp[1] * sY[e*4+2] + vp[2] * sY[e*4+3]);
      }
      acc += m;
    } else {
      int d = comp - 1;
      float m = wr[64 + c] * ss * sY[e * 4 + 1 + d];
      if (srcV) m += wr[192 + c] * srcV[((b * NEL + s) * 64 + c) * 3 + d] * Y0;
      acc += m;
    }
  }
  if (comp == 0) atomicAdd(&catS[(b * NEL + curT) * 320 + offS + c], acc);
  else atomicAdd(&catV[((b * NEL + curT) * 320 + offV + c) * 3 + (comp - 1)], acc);
}

// ---------------------------------------------------------------------------
__global__ void prep_cat_k(float* __restrict__ catS, float* __restrict__ catV,
                           const float* __restrict__ sIn, const float* __restrict__ vIn,
                           int sdim, int vdim)
{
  int i = blockIdx.x * blockDim.x + threadIdx.x;
  if (i >= B_ * NEL * 320) return;
  int col = i % 320, row = i / 320;
  catS[i] = (col < sdim) ? sIn[row * sdim + col] : 0.f;
  #pragma unroll
  for (int d = 0; d < 3; d++)
    catV[i * 3 + d] = (vIn != nullptr && col < vdim) ? vIn[(row * vdim + col) * 3 + d] : 0.f;
}

__global__ void ion_gemm_k(const float* __restrict__ F, const float* __restrict__ W,
                           float* __restrict__ out)
{
  int i = blockIdx.x * 256 + threadIdx.x;
  if (i >= NION * 64) return;
  int row = i >> 6, col = i & 63;
  float acc = 0.f;
  #pragma unroll
  for (int k = 0; k < 32; k++) acc += F[row * 32 + k] * W[k * 64 + col];
  out[i] = acc;
}

__global__ void epilogue_k(const float* __restrict__ ns, const float* __restrict__ nv,
                           const float* __restrict__ sRes, const float* __restrict__ vRes,
                           float* __restrict__ sOut, float* __restrict__ vOut,
                           int hasV, int resid)
{
  int i = blockIdx.x * blockDim.x + threadIdx.x;
  if (i >= B_ * NEL * HCH) return;
  float s = geluf(ns[i]);
  if (resid) s += sRes[i];
  sOut[i] = s;
  if (hasV) {
    float a = nv[3*i], bv = nv[3*i+1], cv = nv[3*i+2];
    float nn = sqrtf(a*a + bv*bv + cv*cv + 1e-12f);
    float sc = geluf(nn) / nn;
    a *= sc; bv *= sc; cv *= sc;
    if (resid) { a += vRes[3*i]; bv += vRes[3*i+1]; cv += vRes[3*i+2]; }
    vOut[3*i] = a; vOut[3*i+1] = bv; vOut[3*i+2] = cv;
  }
}

// ---------------------------------------------------------------------------
static void launch_gemm(const float* A, int ldA, int sA, const float* W, int K, int N,
                        float* D, int ldD, int sD, hipStream_t st)
{
  dim3 g((B_ * NEL) / 16, N / 64);
  gemm_wmma_k<<<g, 128, 0, st>>>(A, ldA, sA, W, N, D, ldD, sD, K);
}

struct MlpP { const float* W[3]; const float* b[3]; };

extern "C" void kernel_launch(void* const* d_in, const int* in_sizes, int n_in,
                              void* d_out, int out_size, void* d_ws, size_t ws_size,
                              hipStream_t stream)
{
  (void)in_sizes; (void)n_in; (void)out_size; (void)ws_size;
  auto F = [&](int i) -> const float* { return (const float*)d_in[i]; };

  const float* diff_ee = F(0);
  const float* dist_ee = F(1);
  const float* diff_ei = F(2);
  const float* dist_ei = F(3);
  const float* feat_el = F(4);
  const float* feat_ee = F(5);
  const float* feat_ei = F(6);
  const float* feat_ion = F(7);

  // params in JAX pytree leaf order: h1, ion, one_el, rad_diff, rad_ion, rad_same
  int p = 8;
  const float *h1Ws[3], *h1Wv[3] = {nullptr, nullptr, nullptr};
  h1Ws[0] = F(p++);
  h1Ws[1] = F(p++); h1Wv[1] = F(p++);
  h1Ws[2] = F(p++); h1Wv[2] = F(p++);
  const float* ionW[3]; for (int n = 0; n < 3; n++) ionW[n] = F(p++);
  const float *oeWs[3], *oeWv[3] = {nullptr, nullptr, nullptr};
  oeWs[0] = F(p++); oeWv[0] = F(p++);
  oeWs[1] = F(p++); oeWv[1] = F(p++);
  oeWs[2] = F(p++);
  MlpP rd[3], ri[3], rs[3];
  auto readMlp = [&](MlpP* arr) {
    for (int n = 0; n < 3; n++) {
      for (int i = 0; i < 3; i++) arr[n].W[i] = F(p++);
      for (int i = 0; i < 3; i++) arr[n].b[i] = F(p++);
    }
  };
  readMlp(rd); readMlp(ri); readMlp(rs);
  const int* same_src = (const int*)d_in[p++];
  const int* same_tgt = (const int*)d_in[p++];
  const int* diff_src = (const int*)d_in[p++];
  const int* diff_tgt = (const int*)d_in[p++];
  const int* ion_src  = (const int*)d_in[p++];
  const int* ion_tgt  = (const int*)d_in[p++];

  // workspace carve-up (~50 MB of fp32)
  float* w = (float*)d_ws;
  auto take = [&](size_t n) { float* q = w; w += n; return q; };
  float* catS  = take((size_t)B_ * NEL * 320);
  float* catV  = take((size_t)B_ * NEL * 320 * 3);
  float* sPing = take((size_t)B_ * NEL * HCH);
  float* sPong = take((size_t)B_ * NEL * HCH);
  float* vPing = take((size_t)B_ * NEL * HCH * 3);
  float* vPong = take((size_t)B_ * NEL * HCH * 3);
  float* h1sB  = take((size_t)B_ * NEL * 64);
  float* h1vB  = take((size_t)B_ * NEL * 64 * 3);
  float* nsB   = take((size_t)B_ * NEL * HCH);
  float* nvB   = take((size_t)B_ * NEL * HCH * 3);
  float* ionsB = take((size_t)NION * 64);

  const float* sIn = feat_el; int sdim = 64;
  const float* vIn = nullptr; int vdim = 0;
  float* sOuts[3] = { sPing, sPong, (float*)d_out };
  float* vOuts[3] = { vPing, vPong, nullptr };

  for (int n = 0; n < 3; n++) {
    const bool hasVout = (n < 2);
    const bool resid   = (n == 1);
    const int  nwEl    = (vdim == 0) ? 128 : 256;
    const int  in_s    = sdim + 192;
    const int  in_v    = vdim + 192;

    { int tot = B_ * NEL * 320;
      prep_cat_k<<<(tot + 255) / 256, 256, 0, stream>>>(catS, catV, sIn, vIn, sdim, vdim); }

    launch_gemm(sIn, sdim, 1, h1Ws[n], sdim, 64, h1sB, 64, 1, stream);
    if (vdim)
      for (int d = 0; d < 3; d++)
        launch_gemm(vIn + d, vdim * 3, 3, h1Wv[n], vdim, 64, h1vB + d, 192, 3, stream);

    ion_gemm_k<<<2, 256, 0, stream>>>(feat_ion, ionW[n], ionsB);

    const float* hv = vdim ? h1vB : nullptr;
    radint_k<<<dim3(480 / 32, B_), 256, 0, stream>>>(
        feat_ee, 32, diff_ee, dist_ee, same_src, same_tgt,
        rs[n].W[0], rs[n].b[0], rs[n].W[1], rs[n].b[1], rs[n].W[2], rs[n].b[2], nwEl,
        h1sB, 1, hv, catS, sdim, catV, vdim);
    radint_k<<<dim3(512 / 32, B_), 256, 0, stream>>>(
        feat_ee, 32, diff_ee, dist_ee, diff_src, diff_tgt,
        rd[n].W[0], rd[n].b[0], rd[n].W[1], rd[n].b[1], rd[n].W[2], rd[n].b[2], nwEl,
        h1sB, 1, hv, catS, sdim + 64, catV, vdim + 64);
    radint_k<<<dim3(256 / 32, B_), 256, 0, stream>>>(
        feat_ei, 8, diff_ei, dist_ei, ion_src, ion_tgt,
        ri[n].W[0], ri[n].b[0], ri[n].W[1], ri[n].b[1], ri[n].W[2], ri[n].b[2], 128,
        ionsB, 0, nullptr, catS, sdim + 128, catV, vdim + 128);

    launch_gemm(catS, 320, 1, oeWs[n], in_s, 128, nsB, 128, 1, stream);
    if (hasVout)
      for (int d = 0; d < 3; d++)
        launch_gemm(catV + d, 960, 3, oeWv[n], in_v, 128, nvB + d, 384, 3, stream);

    { int tot = B_ * NEL * HCH;
      epilogue_k<<<(tot + 255) / 256, 256, 0, stream>>>(
          nsB, nvB, sIn, vIn, sOuts[n], vOuts[n], hasVout ? 1 : 0, resid ? 1 : 0); }

    sIn = sOuts[n]; vIn = hasVout ? vOuts[n] : nullptr;
    sdim = 128; vdim = hasVout ? 128 : 0;
  }
}